// SparseAttention_60370060313380
// MI455X (gfx1250) — compile-verified
//
#include <hip/hip_runtime.h>
#include <hip/hip_bf16.h>
#include <math.h>

// ---------------------------------------------------------------------------
// SparseAttention (NSA-style) for MI455X / gfx1250, wave32 + WMMA f32 path.
// All dense GEMMs run through V_WMMA_F32_16X16X4_F32 (f32-exact tensor op).
// GEMM: each wave owns a 16x32 output tile (2 accumulators) so the A fragment
// is reused across two WMMAs -> 3 LDS dwords per WMMA instead of 4.
// The narrow-N predicates are hoisted out of the K loop so the hot path is a
// branch-free ds_load + 2xWMMA stream the scheduler can pipeline.
// ---------------------------------------------------------------------------

typedef __attribute__((ext_vector_type(2))) float v2f;
typedef __attribute__((ext_vector_type(8))) float v8f;

#define HEADS    16
#define KV_HEADS 4
#define G        4
#define DH       64
#define NSEQ     1024
#define DIM      1024
#define BS       32
#define WBLK     32          // N / BS
#define NSEL     8
#define QKV_LD   1536        // (HEADS + 2*KV)*DH
#define KOFF     1024        // k column offset in qkv
#define VOFF     1280        // v column offset in qkv
#define CDIM     2048
#define HID      2048
#define SCALE    0.125f      // DH^-0.5
#define NEG_MAX  (-3.402823466e38f)

// ---------------- wave helpers (wave32) ----------------
__device__ __forceinline__ float wave_max32(float v) {
    #pragma unroll
    for (int o = 16; o; o >>= 1) v = fmaxf(v, __shfl_xor(v, o, 32));
    return v;
}
__device__ __forceinline__ float wave_sum32(float v) {
    #pragma unroll
    for (int o = 16; o; o >>= 1) v += __shfl_xor(v, o, 32);
    return v;
}
__device__ __forceinline__ float dot64(const float* __restrict__ a,
                                       const float* __restrict__ b) {
    float s = 0.f;
    #pragma unroll
    for (int t = 0; t < 16; ++t) {
        float4 x = ((const float4*)a)[t];
        float4 y = ((const float4*)b)[t];
        s += x.x * y.x + x.y * y.y + x.z * y.z + x.w * y.w;
    }
    return s;
}

// ---------------- 1. RMSNorm ----------------
__global__ void nsa_rmsnorm(const float* __restrict__ inp,
                            const float* __restrict__ g,
                            float* __restrict__ x) {
    __shared__ float red[8];
    __shared__ float rinv;
    const int n = blockIdx.x, tid = threadIdx.x;
    const float* row = inp + (long long)n * DIM;
    float s = 0.f;
    for (int c = tid; c < DIM; c += 256) { float v = row[c]; s += v * v; }
    s = wave_sum32(s);
    if ((tid & 31) == 0) red[tid >> 5] = s;
    __syncthreads();
    if (tid == 0) {
        float t = 0.f;
        #pragma unroll
        for (int i = 0; i < 8; ++i) t += red[i];
        rinv = rsqrtf(t / (float)DIM + 1e-6f);
    }
    __syncthreads();
    const float ri = rinv;
    for (int c = tid; c < DIM; c += 256)
        x[(long long)n * DIM + c] = row[c] * ri * g[c];
}

// ---------------- 2. generic WMMA-f32 GEMM ----------------
// C[z] = act(A[z] @ B + bias).  Block = 128 threads (4 waves).
// Block tile: 16 rows x 128 cols; each wave owns a 16x32 tile (2 WMMA tiles,
// A fragment shared between them).  A/B staged through LDS per 64-wide K chunk.
// M multiple of 16, N multiple of 16, K multiple of 64 (true for all uses).
__global__ void nsa_wmma_gemm(const float* __restrict__ A,
                              const float* __restrict__ B,
                              const float* __restrict__ bias,
                              float* __restrict__ C,
                              int M, int N, int K, int lda, int ldb, int ldc,
                              long long strideA, long long strideC,
                              int act, int has_bias) {
    __shared__ float As[16][68];
    __shared__ float Bs[64][132];

    const int m0 = blockIdx.x * 16;
    const int n0 = blockIdx.y * 128;
    const int z  = blockIdx.z;
    const float* Ab = A + (long long)z * strideA;
    float*       Cb = C + (long long)z * strideC;

    const int tid  = threadIdx.x;
    const int lane = tid & 31;
    const int wv   = tid >> 5;               // 0..3
    const int cw   = lane & 15;              // col-in-subtile / A row
    const int kHalf  = (lane >> 4) << 1;     // 0 or 2 (K split across lane halves)
    const int rowOff = (lane >> 4) << 3;     // 0 or 8 (C/D row split)
    const int wcol0 = wv * 32 + cw;          // first subtile col within block tile
    const int wcol1 = wcol0 + 16;            // second subtile col
    const bool act0 = (n0 + wv * 32) < N;
    const bool act1 = (n0 + wv * 32 + 16) < N;

    v8f acc0 = {};
    v8f acc1 = {};
    for (int k0 = 0; k0 < K; k0 += 64) {
        // cooperative A tile: 16 x 64  (1024 elems / 128 threads)
        for (int t = tid; t < 16 * 64; t += 128) {
            int r = t >> 6, c = t & 63;
            As[r][c] = Ab[(long long)(m0 + r) * lda + (k0 + c)];
        }
        // cooperative B tile: 64 x 128 (guard N for narrow outputs)
        for (int t = tid; t < 64 * 128; t += 128) {
            int r = t >> 7, c = t & 127;
            int col = n0 + c;
            Bs[r][c] = (col < N) ? B[(long long)(k0 + r) * ldb + col] : 0.0f;
        }
        // prefetch next K chunk while this one computes (global_prefetch_b8)
        if (k0 + 64 < K) {
            const float* pa = &Ab[(long long)(m0 + (tid & 15)) * lda + (k0 + 64) + ((tid >> 4) << 3)];
            __builtin_prefetch(pa, 0, 1);
            int pcol = n0 + ((tid & 31) << 2);
            if (pcol < N) {
                const float* pb = &B[(long long)(k0 + 64 + (tid >> 5) * 16) * ldb + pcol];
                __builtin_prefetch(pb, 0, 1);
            }
        }
        __syncthreads();
        if (act0 & act1) {
            // hot path: branch-free, 2 WMMAs per A fragment
            #pragma unroll
            for (int kk = 0; kk < 64; kk += 4) {
                v2f a, b0, b1;
                a.x  = As[cw][kk + kHalf];
                a.y  = As[cw][kk + kHalf + 1];
                b0.x = Bs[kk + kHalf][wcol0];
                b0.y = Bs[kk + kHalf + 1][wcol0];
                b1.x = Bs[kk + kHalf][wcol1];
                b1.y = Bs[kk + kHalf + 1][wcol1];
                acc0 = __builtin_amdgcn_wmma_f32_16x16x4_f32(
                    false, a, false, b0, (short)0, acc0, false, false);
                acc1 = __builtin_amdgcn_wmma_f32_16x16x4_f32(
                    false, a, false, b1, (short)0, acc1, false, false);
            }
        } else if (act0) {
            #pragma unroll
            for (int kk = 0; kk < 64; kk += 4) {
                v2f a, b0;
                a.x  = As[cw][kk + kHalf];
                a.y  = As[cw][kk + kHalf + 1];
                b0.x = Bs[kk + kHalf][wcol0];
                b0.y = Bs[kk + kHalf + 1][wcol0];
                acc0 = __builtin_amdgcn_wmma_f32_16x16x4_f32(
                    false, a, false, b0, (short)0, acc0, false, false);
            }
        }
        __syncthreads();
    }
    if (act0) {
        const int col0 = n0 + wcol0;
        const float bv0 = has_bias ? bias[col0] : 0.0f;
        #pragma unroll
        for (int r = 0; r < 8; ++r) {
            float v = acc0[r] + bv0;
            if (act == 1)      v = fmaxf(v, 0.0f);
            else if (act == 2) v = 1.0f / (1.0f + __expf(-v));
            Cb[(long long)(m0 + rowOff + r) * ldc + col0] = v;
        }
        if (act1) {
            const int col1 = n0 + wcol1;
            const float bv1 = has_bias ? bias[col1] : 0.0f;
            #pragma unroll
            for (int r = 0; r < 8; ++r) {
                float v = acc1[r] + bv1;
                if (act == 1)      v = fmaxf(v, 0.0f);
                else if (act == 2) v = 1.0f / (1.0f + __expf(-v));
                Cb[(long long)(m0 + rowOff + r) * ldc + col1] = v;
            }
        }
    }
}

// ---------------- 3. build kpe/vpe (block-pos added, (KV,W,CDIM)) ----------------
__global__ void nsa_addpos(const float* __restrict__ qkv,
                           const float* __restrict__ kpos,
                           const float* __restrict__ vpos,
                           float* __restrict__ kpe, float* __restrict__ vpe) {
    int tid = blockIdx.x * blockDim.x + threadIdx.x;   // KV*W*CDIM = 262144
    if (tid >= KV_HEADS * WBLK * CDIM) return;
    int d  = tid & 63;
    int bs = (tid >> 6) & 31;
    int w  = (tid >> 11) & 31;
    int h  = tid >> 16;
    int nrow = w * BS + bs;
    long long src = (long long)nrow * QKV_LD + h * DH + d;
    long long o   = ((long long)h * WBLK + w) * CDIM + bs * DH + d;
    kpe[o] = qkv[src + KOFF] + kpos[(h * BS + bs) * DH + d];
    vpe[o] = qkv[src + VOFF] + vpos[(h * BS + bs) * DH + d];
}

// ---------------- 4. mem_kv -> row 0 of ck/cv ----------------
__global__ void nsa_memrow(const float* __restrict__ mem_kv,
                           float* __restrict__ ckf, float* __restrict__ cvf) {
    int tid = blockIdx.x * blockDim.x + threadIdx.x;   // 2*KV*DH = 512
    if (tid >= 2 * KV_HEADS * DH) return;
    int d = tid & 63, h = (tid >> 6) & 3, s = tid >> 8;
    float v = mem_kv[((s * KV_HEADS + h) * 1) * DH + d];
    (s == 0 ? ckf : cvf)[(h * (WBLK + 1)) * DH + d] = v;
}

// ---------------- 5. compressed attention (+ raw csim store) ----------------
// one wave per (head, query).  j-per-lane for QK, d-per-lane for PV.
__global__ void nsa_cattn(const float* __restrict__ qkv,
                          const float* __restrict__ ckf,
                          const float* __restrict__ cvf,
                          float* __restrict__ csim, float* __restrict__ cout) {
    __shared__ float p_sm[8][36];
    const int tid = threadIdx.x, lane = tid & 31, wv = tid >> 5;
    const int idx = blockIdx.x * 8 + wv;        // HEADS*NSEQ pairs
    const int h = idx >> 10, n = idx & 1023;
    const int kvh = h >> 2;
    const float* qrow = qkv + (long long)n * QKV_LD + h * DH;

    float s1 = dot64(qrow, ckf + (kvh * 33 + lane) * DH) * SCALE;
    float s2 = 0.f;
    if (lane == 0) s2 = dot64(qrow, ckf + (kvh * 33 + 32) * DH) * SCALE;
    long long cb = ((long long)h * NSEQ + n) * 33;
    csim[cb + lane] = s1;
    if (lane == 0) csim[cb + 32] = s2;

    float lm = (lane == 0) ? fmaxf(s1, s2) : s1;
    lm = wave_max32(lm);
    float le = __expf(s1 - lm);
    if (lane == 0) le += __expf(s2 - lm);
    float Z = wave_sum32(le);
    p_sm[wv][lane] = __expf(s1 - lm) / Z;
    if (lane == 0) p_sm[wv][32] = __expf(s2 - lm) / Z;
    __syncthreads();

    float a0 = 0.f, a1 = 0.f;
    #pragma unroll 4
    for (int j = 0; j < 33; ++j) {
        const float* vr = cvf + (kvh * 33 + j) * DH;
        float pw = p_sm[wv][j];
        a0 += pw * vr[lane];
        a1 += pw * vr[lane + 32];
    }
    long long ob = ((long long)h * NSEQ + n) * DH;
    cout[ob + lane] = a0;
    cout[ob + lane + 32] = a1;
}

// ---------------- 6. interleaved rotary ----------------
__global__ void nsa_rotary(const float* __restrict__ src, int rowStride, int colOff,
                           int nHeads, float* __restrict__ dst) {
    int tid = blockIdx.x * blockDim.x + threadIdx.x;
    if (tid >= nHeads * NSEQ * 32) return;
    int i = tid & 31;                 // pair index
    int n = (tid >> 5) & 1023;
    int h = tid >> 15;
    float inv = __expf(-((float)(2 * i) / 64.0f) * 9.2103403719761836f); // 10000^{-2i/64}
    float ang = (float)n * inv;
    float s, c;
    sincosf(ang, &s, &c);
    const float* sp = src + (long long)n * rowStride + colOff + h * DH + 2 * i;
    float x0 = sp[0], x1 = sp[1];
    long long ob = ((long long)h * NSEQ + n) * DH + 2 * i;
    dst[ob]     = x0 * c - x1 * s;
    dst[ob + 1] = x1 * c + x0 * s;
}

// ---------------- 7. block selection (top-8 + own block) ----------------
__global__ void nsa_select(const float* __restrict__ csim,
                           int* __restrict__ selIdx, int* __restrict__ selMask) {
    const int tid = threadIdx.x, lane = tid & 31, wv = tid >> 5;
    const int idx = blockIdx.x * 8 + wv;         // KV*NSEQ pairs
    const int kvh = idx >> 10, n = idx & 1023;

    float v = 0.f;
    #pragma unroll
    for (int g = 0; g < G; ++g)
        v += csim[(((long long)(kvh * G + g)) * NSEQ + n) * 33 + 1 + lane];
    v *= 0.25f;

    // softmax incl. the -1000 pad slot, then probabilities per block
    float m = fmaxf(wave_max32(v), -1000.0f);
    float e = __expf(v - m);
    float Z = wave_sum32(e) + __expf(-1000.0f - m);
    float prob = e / Z;

    float cur = v;
    int* ob = selIdx + ((long long)kvh * NSEQ + n) * (NSEL + 1);
    int* mb = selMask + ((long long)kvh * NSEQ + n) * (NSEL + 1);
    for (int s = 0; s < NSEL; ++s) {
        float bv = cur; int bi = lane;
        #pragma unroll
        for (int o = 16; o; o >>= 1) {
            float ov = __shfl_xor(bv, o, 32);
            int   oi = __shfl_xor(bi, o, 32);
            if (ov > bv || (ov == bv && oi < bi)) { bv = ov; bi = oi; }
        }
        float wprob = __shfl(prob, bi, 32);
        if (lane == 0) { ob[s] = bi; mb[s] = (wprob > 1e-10f) ? 1 : 0; }
        if (lane == bi) cur = NEG_MAX;
    }
    if (lane == 0) { ob[NSEL] = n >> 5; mb[NSEL] = 1; }
}

// ---------------- 8. fine (selected-block) attention ----------------
// one wave per (head, query).  lane = row within selected block (coalesced K),
// PV pass is d-per-lane (coalesced V) with probabilities staged in LDS.
__global__ void nsa_fattn(const float* __restrict__ qr,
                          const float* __restrict__ kr,
                          const float* __restrict__ qkv,
                          const int* __restrict__ selIdx,
                          const int* __restrict__ selMask,
                          float* __restrict__ fout) {
    __shared__ float p_sm[8][292];
    const int tid = threadIdx.x, lane = tid & 31, wv = tid >> 5;
    const int idx = blockIdx.x * 8 + wv;
    const int h = idx >> 10, n = idx & 1023;
    const int kvh = h >> 2;

    const float* qrow = qr + ((long long)h * NSEQ + n) * DH;
    float4 qv[16];
    #pragma unroll
    for (int t = 0; t < 16; ++t) qv[t] = ((const float4*)qrow)[t];

    const int* sb = selIdx + ((long long)kvh * NSEQ + n) * (NSEL + 1);
    const int* mb = selMask + ((long long)kvh * NSEQ + n) * (NSEL + 1);

    int blk[NSEL + 1];
    float sims[NSEL + 1];
    #pragma unroll
    for (int i = 0; i <= NSEL; ++i) {
        blk[i] = sb[i];
        const float* krow = kr + ((long long)kvh * NSEQ + blk[i] * BS + lane) * DH;
        float s = 0.f;
        #pragma unroll
        for (int t = 0; t < 16; ++t) {
            float4 y = ((const float4*)krow)[t];
            s += qv[t].x * y.x + qv[t].y * y.y + qv[t].z * y.z + qv[t].w * y.w;
        }
        s *= SCALE;
        sims[i] = mb[i] ? s : NEG_MAX;
    }
    float lm = sims[0];
    #pragma unroll
    for (int i = 1; i <= NSEL; ++i) lm = fmaxf(lm, sims[i]);
    lm = wave_max32(lm);
    float le = 0.f;
    #pragma unroll
    for (int i = 0; i <= NSEL; ++i) le += __expf(sims[i] - lm);
    float Z = wave_sum32(le);
    #pragma unroll
    for (int i = 0; i <= NSEL; ++i)
        p_sm[wv][i * 32 + lane] = __expf(sims[i] - lm) / Z;
    __syncthreads();

    float a0 = 0.f, a1 = 0.f;
    #pragma unroll
    for (int i = 0; i <= NSEL; ++i) {
        long long rowb = (long long)blk[i] * BS;
        for (int r = 0; r < BS; ++r) {
            float pw = p_sm[wv][i * 32 + r];
            const float* vr = qkv + (rowb + r) * QKV_LD + VOFF + kvh * DH;
            a0 += pw * vr[lane];
            a1 += pw * vr[lane + 32];
        }
    }
    long long ob = ((long long)h * NSEQ + n) * DH;
    fout[ob + lane] = a0;
    fout[ob + lane + 32] = a1;
}

// ---------------- 9. gated mix of branches ----------------
__global__ void nsa_mix(const float* __restrict__ cout,
                        const float* __restrict__ fout,
                        const float* __restrict__ gates,
                        float* __restrict__ att) {
    int tid = blockIdx.x * blockDim.x + threadIdx.x;    // NSEQ*DIM
    if (tid >= NSEQ * DIM) return;
    int d = tid & 63, h = (tid >> 6) & 15, n = tid >> 10;
    float g0 = gates[n * 32 + h * 2];
    float g1 = gates[n * 32 + h * 2 + 1];
    long long src = ((long long)h * NSEQ + n) * DH + d;
    att[tid] = g0 * cout[src] + g1 * fout[src];
}

// ---------------------------------------------------------------------------
extern "C" void kernel_launch(void* const* d_in, const int* in_sizes, int n_in,
                              void* d_out, int out_size, void* d_ws, size_t ws_size,
                              hipStream_t stream) {
    const float* inp    = (const float*)d_in[0];
    const float* g_norm = (const float*)d_in[1];
    const float* w_qkv  = (const float*)d_in[2];
    const float* mem_kv = (const float*)d_in[3];
    const float* k_pos  = (const float*)d_in[4];
    const float* v_pos  = (const float*)d_in[5];
    const float* k_w1   = (const float*)d_in[6];
    const float* k_b1   = (const float*)d_in[7];
    const float* k_w2   = (const float*)d_in[8];
    const float* k_b2   = (const float*)d_in[9];
    const float* v_w1   = (const float*)d_in[10];
    const float* v_b1   = (const float*)d_in[11];
    const float* v_w2   = (const float*)d_in[12];
    const float* v_b2   = (const float*)d_in[13];
    const float* gate_w = (const float*)d_in[14];
    const float* gate_b = (const float*)d_in[15];
    const float* w_out  = (const float*)d_in[16];
    float* out = (float*)d_out;

    // workspace layout (floats); total ~8.79M floats ~= 34 MB
    float* ws   = (float*)d_ws;
    float* X    = ws;                    // 1,048,576
    float* QKV  = X    + 1048576;        // 1,572,864
    float* KPE  = QKV  + 1572864;        //   262,144
    float* VPE  = KPE  + 262144;         //   262,144
    float* HK   = VPE  + 262144;         //   262,144
    float* HV   = HK   + 262144;         //   262,144
    float* CKF  = HV   + 262144;         //     8,448   (KV,33,DH)
    float* CVF  = CKF  + 8448;           //     8,448
    float* CSIM = CVF  + 8448;           //   540,672   (H,N,33)
    float* COUT = CSIM + 540672;         // 1,048,576
    float* QR   = COUT + 1048576;        // 1,048,576
    float* KR   = QR   + 1048576;        //   262,144
    float* FOUT = KR   + 262144;         // 1,048,576
    float* GATES= FOUT + 1048576;        //    32,768
    float* ATT  = GATES+ 32768;          // 1,048,576
    int*   SELI = (int*)(ATT + 1048576); //    36,864 ints
    int*   SELM = SELI + 36864;          //    36,864 ints

    const dim3 blk(256);
    const dim3 gblk(128);                // 4 waves per GEMM block

    // 1. RMSNorm
    nsa_rmsnorm<<<dim3(NSEQ), blk, 0, stream>>>(inp, g_norm, X);

    // 2. qkv = x @ w_qkv     (1024 x 1536 x 1024)
    nsa_wmma_gemm<<<dim3(64, 12, 1), gblk, 0, stream>>>(
        X, w_qkv, nullptr, QKV, NSEQ, QKV_LD, DIM, DIM, QKV_LD, QKV_LD,
        0, 0, /*act*/0, /*bias*/0);

    // 3. gates = sigmoid(x @ gate_w + gate_b)   (1024 x 32 x 1024)
    nsa_wmma_gemm<<<dim3(64, 1, 1), gblk, 0, stream>>>(
        X, gate_w, gate_b, GATES, NSEQ, 32, DIM, DIM, 32, 32,
        0, 0, /*act*/2, /*bias*/1);

    // 4. kpe/vpe
    nsa_addpos<<<dim3(1024), blk, 0, stream>>>(QKV, k_pos, v_pos, KPE, VPE);

    // 5. compress MLP layer 1 (relu), batched over KV heads: (32 x 2048 x 2048)
    nsa_wmma_gemm<<<dim3(2, 16, KV_HEADS), gblk, 0, stream>>>(
        KPE, k_w1, k_b1, HK, WBLK, HID, CDIM, CDIM, HID, HID,
        (long long)WBLK * CDIM, (long long)WBLK * HID, 1, 1);
    nsa_wmma_gemm<<<dim3(2, 16, KV_HEADS), gblk, 0, stream>>>(
        VPE, v_w1, v_b1, HV, WBLK, HID, CDIM, CDIM, HID, HID,
        (long long)WBLK * CDIM, (long long)WBLK * HID, 1, 1);

    // 6. mem_kv into row 0 of ck/cv
    nsa_memrow<<<dim3(2), blk, 0, stream>>>(mem_kv, CKF, CVF);

    // 7. compress MLP layer 2: (32 x 64 x 2048), writes rows 1..32 of ck/cv
    nsa_wmma_gemm<<<dim3(2, 1, KV_HEADS), gblk, 0, stream>>>(
        HK, k_w2, k_b2, CKF + DH, WBLK, DH, HID, HID, DH, DH,
        (long long)WBLK * HID, (long long)(WBLK + 1) * DH, 0, 1);
    nsa_wmma_gemm<<<dim3(2, 1, KV_HEADS), gblk, 0, stream>>>(
        HV, v_w2, v_b2, CVF + DH, WBLK, DH, HID, HID, DH, DH,
        (long long)WBLK * HID, (long long)(WBLK + 1) * DH, 0, 1);

    // 8. compressed attention (stores raw csim for selection)
    nsa_cattn<<<dim3(HEADS * NSEQ / 8), blk, 0, stream>>>(QKV, CKF, CVF, CSIM, COUT);

    // 9. rotary on q (16 heads) and k (4 heads)
    nsa_rotary<<<dim3(HEADS * NSEQ * 32 / 256), blk, 0, stream>>>(QKV, QKV_LD, 0, HEADS, QR);
    nsa_rotary<<<dim3(KV_HEADS * NSEQ * 32 / 256), blk, 0, stream>>>(QKV, QKV_LD, KOFF, KV_HEADS, KR);

    // 10. top-8 block selection + own block
    nsa_select<<<dim3(KV_HEADS * NSEQ / 8), blk, 0, stream>>>(CSIM, SELI, SELM);

    // 11. fine attention over gathered blocks
    nsa_fattn<<<dim3(HEADS * NSEQ / 8), blk, 0, stream>>>(QR, KR, QKV, SELI, SELM, FOUT);

    // 12. gated mix -> (N, H*DH)
    nsa_mix<<<dim3(NSEQ * DIM / 256), blk, 0, stream>>>(COUT, FOUT, GATES, ATT);

    // 13. out = att @ w_out   (1024 x 1024 x 1024)
    nsa_wmma_gemm<<<dim3(64, 8, 1), gblk, 0, stream>>>(
        ATT, w_out, nullptr, out, NSEQ, DIM, DIM, DIM, DIM, DIM,
        0, 0, 0, 0);
}